// LinearTransferFeatures_40553081209242
// MI455X (gfx1250) — compile-verified
//
#include <hip/hip_runtime.h>
#include <hip/hip_bf16.h>

#define NTOK   8192          // T*B = 256*32
#define DEMB   128
#define NEXP   9             // MULTI_INFER_NUM + 1
#define GROUPS 512           // NTOK / 16 (max groups any expert can have)

typedef __attribute__((ext_vector_type(2))) float v2f;
typedef __attribute__((ext_vector_type(8))) float v8f;

// ---------------------------------------------------------------------------
// Kernel 0: zero the per-expert counters (workspace is not re-poisoned
// between replays, so we must re-init every launch).
// ---------------------------------------------------------------------------
__global__ void ltf_zero_counts(int* __restrict__ counts) {
    if (threadIdx.x < NEXP) counts[threadIdx.x] = 0;
}

// ---------------------------------------------------------------------------
// Kernel 1: bin token ids by expert. e = min(pos, 8). Token ids fit in u16.
// ---------------------------------------------------------------------------
__global__ void ltf_scatter(const long long* __restrict__ positions,
                            int* __restrict__ counts,
                            unsigned short* __restrict__ lists) {
    const int i = blockIdx.x * blockDim.x + threadIdx.x;
    if (i >= NTOK) return;
    const long long p = positions[i];
    const int e = (p < 8) ? (int)p : 8;
    const int idx = atomicAdd(&counts[e], 1);
    lists[e * NTOK + idx] = (unsigned short)i;
}

// ---------------------------------------------------------------------------
// Kernel 2: per (expert, group-of-16-tokens) block.
//   A (16x128)  = gathered token feature rows
//   B (128x128) = W[e]  (L2-resident, re-read per block)
//   D (16x128)  = A*B + bias, scattered back to token rows.
// 4 waves per block; wave w computes N-tiles 2w and 2w+1 with
// V_WMMA_F32_16X16X4_F32 accumulating over K = 128 in steps of 4.
// ---------------------------------------------------------------------------
__global__ void ltf_compute(const float* __restrict__ X,      // [NTOK,128]
                            const float* __restrict__ W,      // [9,128,128]
                            const float* __restrict__ bias,   // [9,128]
                            const unsigned short* __restrict__ lists,
                            const int* __restrict__ counts,
                            float* __restrict__ Y) {          // [NTOK,128]
    const int e    = blockIdx.y;
    const int cnt  = counts[e];          // uniform scalar -> uniform branch
    const int base = blockIdx.x * 16;
    if (base >= cnt) return;             // whole block exits together
    const int valid = (cnt - base < 16) ? (cnt - base) : 16;

    const int tid  = threadIdx.x;        // 0..127
    const int wave = tid >> 5;           // 0..3
    const int lane = tid & 31;
    const int half = lane >> 4;          // 0: lanes 0-15, 1: lanes 16-31
    const int nl   = lane & 15;

    // Stage the 16 token ids for this group; pad with a valid id so WMMA
    // operands are well-defined, and mask the padded stores later.
    __shared__ int tok[16];
    const unsigned short* lst = lists + e * NTOK + base;
    if (tid < 16) tok[tid] = (tid < valid) ? (int)lst[tid] : (int)lst[0];
    __syncthreads();                     // EXEC back to all-1s before WMMA

    const float* We = W + e * (DEMB * DEMB);
    const float* be = bias + e * DEMB;

    // A operand row for this lane is M = nl (same for both lane halves).
    const float* xrow = X + tok[nl] * DEMB;

    for (int nt = 0; nt < 2; ++nt) {
        const int n = (wave * 2 + nt) * 16 + nl;   // this lane's output column
        v8f c = {};
        for (int k = 0; k < DEMB; k += 4) {
            // A 16x4 f32: vgpr0 = K {0,2}, vgpr1 = K {1,3} across lane halves
            v2f a = *(const v2f*)(xrow + k + 2 * half);   // 8B-aligned b64 load
            // B 4x16 f32: vgpr0 = row K {0,2}, vgpr1 = row K {1,3}
            v2f b;
            b.x = We[(k + 2 * half) * DEMB + n];
            b.y = We[(k + 2 * half + 1) * DEMB + n];
            c = __builtin_amdgcn_wmma_f32_16x16x4_f32(
                    /*neg_a=*/false, a, /*neg_b=*/false, b,
                    /*c_mod=*/(short)0, c, /*reuse_a=*/false, /*reuse_b=*/false);
        }
        const float bv = be[n];
        // C/D layout: VGPR r -> M = r + 8*half, N = nl
        #pragma unroll
        for (int r = 0; r < 8; ++r) {
            const int m = r + 8 * half;
            if (m < valid) {             // uniform-per-half mask on padded rows
                Y[tok[m] * DEMB + n] = c[r] + bv;
            }
        }
    }
}

extern "C" void kernel_launch(void* const* d_in, const int* in_sizes, int n_in,
                              void* d_out, int out_size, void* d_ws, size_t ws_size,
                              hipStream_t stream) {
    (void)in_sizes; (void)n_in; (void)out_size; (void)ws_size;

    const long long* positions = (const long long*)d_in[0];  // int64 [T,B]
    const float*     outputs   = (const float*)d_in[1];      // f32 [T,B,128]
    const float*     W         = (const float*)d_in[2];      // f32 [9,128,128]
    const float*     bias      = (const float*)d_in[3];      // f32 [9,128]
    float*           Y         = (float*)d_out;              // f32 [T,B,128]

    int*            counts = (int*)d_ws;                                  // 9 ints
    unsigned short* lists  = (unsigned short*)((char*)d_ws + 256);        // 9*8192 u16

    ltf_zero_counts<<<1, 32, 0, stream>>>(counts);
    ltf_scatter<<<NTOK / 256, 256, 0, stream>>>(positions, counts, lists);
    ltf_compute<<<dim3(GROUPS, NEXP), 128, 0, stream>>>(outputs, W, bias, lists, counts, Y);
}